// Pointer_32538672235067
// MI455X (gfx1250) — compile-verified
//
#include <hip/hip_runtime.h>
#include <math.h>

#define BB 512
#define HH 128
#define NN 512

typedef __attribute__((ext_vector_type(2))) float v2f;
typedef __attribute__((ext_vector_type(8))) float v8f;

__device__ __forceinline__ float sigm(float x) { return 1.0f / (1.0f + __expf(-x)); }

// -------------------------------------------------------------------------
// Kernel 1: GRU step (per batch) + per-batch attention bias c_attn = Wa_r @ rnn_out
// -------------------------------------------------------------------------
__global__ void __launch_bounds__(128)
gru_kernel(const float* __restrict__ dec, const float* __restrict__ lhh,
           const float* __restrict__ w_ih, const float* __restrict__ w_hh,
           const float* __restrict__ b_ih, const float* __restrict__ b_hh,
           const float* __restrict__ W_attn,
           float* __restrict__ new_hh, float* __restrict__ c_attn)
{
    __shared__ float sx[HH], sh[HH], so[HH];
    const int b = blockIdx.x, t = threadIdx.x;
    sx[t] = dec[b * HH + t];        // decoder_hidden[b, t, 0]
    sh[t] = lhh[b * HH + t];        // last_hh[0, b, t]
    __syncthreads();

    float gi0 = b_ih[t], gi1 = b_ih[t + HH], gi2 = b_ih[t + 2 * HH];
    float gh0 = b_hh[t], gh1 = b_hh[t + HH], gh2 = b_hh[t + 2 * HH];
    const float* wi0 = &w_ih[(size_t)t * HH];
    const float* wi1 = &w_ih[(size_t)(t + HH) * HH];
    const float* wi2 = &w_ih[(size_t)(t + 2 * HH) * HH];
    const float* wh0 = &w_hh[(size_t)t * HH];
    const float* wh1 = &w_hh[(size_t)(t + HH) * HH];
    const float* wh2 = &w_hh[(size_t)(t + 2 * HH) * HH];
#pragma unroll 8
    for (int k = 0; k < HH; ++k) {
        const float xk = sx[k], hk = sh[k];
        gi0 += xk * wi0[k]; gi1 += xk * wi1[k]; gi2 += xk * wi2[k];
        gh0 += hk * wh0[k]; gh1 += hk * wh1[k]; gh2 += hk * wh2[k];
    }
    const float r = sigm(gi0 + gh0);
    const float z = sigm(gi1 + gh1);
    const float n = tanhf(gi2 + r * gh2);
    const float out = (1.0f - z) * n + z * sh[t];
    new_hh[b * HH + t] = out;
    so[t] = out;
    __syncthreads();

    // c_attn[b,t] = sum_k W_attn[t, 256+k] * rnn_out[b,k]
    float c = 0.f;
    const float* wr = &W_attn[(size_t)t * 384 + 256];
#pragma unroll 8
    for (int k = 0; k < HH; ++k) c += wr[k] * so[k];
    c_attn[b * HH + t] = c;
}

// -------------------------------------------------------------------------
// Kernel 2: E[b,n] = v_attn . tanh(Wa_s @ s_n + Wa_d @ d_n + c_attn[b])
// Block: one batch x 128 columns; 4 waves, each owns TWO 16-col N-tiles.
// k-outer / m-inner: 16 independent accumulator chains, B frags reused
// across M-tiles, each A (weight) fragment feeds 2 WMMAs.
// -------------------------------------------------------------------------
__global__ void __launch_bounds__(128)
attn_scores_kernel(const float* __restrict__ S, const float* __restrict__ D,
                   const float* __restrict__ W_attn, const float* __restrict__ v_attn,
                   const float* __restrict__ c_attn, float* __restrict__ E)
{
    __shared__ __align__(16) float sS[32 * 128];
    __shared__ __align__(16) float sD[32 * 128];
    __shared__ float sC[HH], sV[HH];

    const int b  = blockIdx.y;
    const int n0 = blockIdx.x * 128;
    const int t  = threadIdx.x;
    const int lane = t & 31;
    const int wave = t >> 5;
    const int l15 = lane & 15;
    const int nl0 = (wave << 4) + l15;   // first owned column (local)
    const int nl1 = nl0 + 64;            // second owned column (local)
    const bool hi = lane >= 16;
    const int kofs = hi ? 2 : 0;

    sC[t] = c_attn[b * HH + t];
    sV[t] = v_attn[t];
    const size_t sb = (size_t)b * HH * NN;

    v8f acc[2][8];
#pragma unroll
    for (int q = 0; q < 2; ++q)
#pragma unroll
        for (int m = 0; m < 8; ++m)
            acc[q][m] = (v8f){0.f,0.f,0.f,0.f,0.f,0.f,0.f,0.f};

    for (int stage = 0; stage < 4; ++stage) {
        const int kbase = stage * 32;
        __syncthreads();
        // stage 32 K-rows x 128 cols of S and D into LDS (float4, coalesced)
#pragma unroll
        for (int i = 0; i < 8; ++i) {
            const int e4 = i * 128 + t;      // 0..1023
            const int k  = e4 >> 5;          // row (32 float4 per row)
            const int j  = e4 & 31;
            ((float4*)sS)[e4] = *(const float4*)(S + sb + (size_t)(kbase + k) * NN + n0 + j * 4);
            ((float4*)sD)[e4] = *(const float4*)(D + sb + (size_t)(kbase + k) * NN + n0 + j * 4);
        }
        __syncthreads();

        const float* Wk = W_attn + (size_t)l15 * 384 + kbase + kofs;
        for (int k0 = 0; k0 < 32; k0 += 4) {
            v2f bS0, bS1, bD0, bD1;
            bS0.x = sS[(k0 + kofs) * 128 + nl0]; bS0.y = sS[(k0 + kofs + 1) * 128 + nl0];
            bS1.x = sS[(k0 + kofs) * 128 + nl1]; bS1.y = sS[(k0 + kofs + 1) * 128 + nl1];
            bD0.x = sD[(k0 + kofs) * 128 + nl0]; bD0.y = sD[(k0 + kofs + 1) * 128 + nl0];
            bD1.x = sD[(k0 + kofs) * 128 + nl1]; bD1.y = sD[(k0 + kofs + 1) * 128 + nl1];
#pragma unroll
            for (int m = 0; m < 8; ++m) {
                const float* Arow = Wk + (size_t)(m * 16) * 384 + k0;
                v2f aS; aS.x = Arow[0];   aS.y = Arow[1];
                v2f aD; aD.x = Arow[128]; aD.y = Arow[129];
                acc[0][m] = __builtin_amdgcn_wmma_f32_16x16x4_f32(
                    false, aS, false, bS0, (short)0, acc[0][m], false, false);
                acc[1][m] = __builtin_amdgcn_wmma_f32_16x16x4_f32(
                    false, aS, false, bS1, (short)0, acc[1][m], false, false);
                acc[0][m] = __builtin_amdgcn_wmma_f32_16x16x4_f32(
                    false, aD, false, bD0, (short)0, acc[0][m], false, false);
                acc[1][m] = __builtin_amdgcn_wmma_f32_16x16x4_f32(
                    false, aD, false, bD1, (short)0, acc[1][m], false, false);
            }
        }
    }

    // epilogue: bias + tanh + v_attn dot, reduce rows within lane, then halves
    float e0 = 0.f, e1 = 0.f;
#pragma unroll
    for (int m = 0; m < 8; ++m) {
        const int rbase = m * 16 + (hi ? 8 : 0);
#pragma unroll
        for (int v = 0; v < 8; ++v) {
            const int row = rbase + v;
            const float w = sV[row], cb = sC[row];
            e0 += w * tanhf(acc[0][m][v] + cb);
            e1 += w * tanhf(acc[1][m][v] + cb);
        }
    }
    e0 += __shfl_xor(e0, 16, 32);
    e1 += __shfl_xor(e1, 16, 32);
    if (!hi) {
        E[(size_t)b * NN + n0 + nl0] = e0;
        E[(size_t)b * NN + n0 + nl1] = e1;
    }
}

// -------------------------------------------------------------------------
// Kernel 3: in-place softmax over N per batch
// -------------------------------------------------------------------------
__global__ void __launch_bounds__(128)
softmax_kernel(float* __restrict__ E)
{
    __shared__ float red[128];
    const int b = blockIdx.x, t = threadIdx.x;
    float* e = E + (size_t)b * NN;
    float x0 = e[t], x1 = e[t + 128], x2 = e[t + 256], x3 = e[t + 384];
    float m = fmaxf(fmaxf(x0, x1), fmaxf(x2, x3));
    red[t] = m; __syncthreads();
    for (int s = 64; s > 0; s >>= 1) { if (t < s) red[t] = fmaxf(red[t], red[t + s]); __syncthreads(); }
    m = red[0]; __syncthreads();
    x0 = __expf(x0 - m); x1 = __expf(x1 - m); x2 = __expf(x2 - m); x3 = __expf(x3 - m);
    red[t] = x0 + x1 + x2 + x3; __syncthreads();
    for (int s = 64; s > 0; s >>= 1) { if (t < s) red[t] += red[t + s]; __syncthreads(); }
    const float inv = 1.0f / red[0];
    e[t] = x0 * inv; e[t + 128] = x1 * inv; e[t + 256] = x2 * inv; e[t + 384] = x3 * inv;
}

// -------------------------------------------------------------------------
// Kernel 4: context[b,h] = sum_n attns[b,n]*static[b,h,n]; c_ptr = Wp_c @ context
// -------------------------------------------------------------------------
__global__ void __launch_bounds__(128)
context_kernel(const float* __restrict__ S, const float* __restrict__ attns,
               const float* __restrict__ W_ptr, float* __restrict__ c_ptr)
{
    __shared__ float sA[NN];
    __shared__ float sctx[HH];
    const int b = blockIdx.x, t = threadIdx.x;
    const int lane = t & 31, wave = t >> 5;
    for (int i = t; i < NN; i += 128) sA[i] = attns[(size_t)b * NN + i];
    __syncthreads();

    for (int h = wave * 32; h < wave * 32 + 32; ++h) {
        const float* row = S + (size_t)b * HH * NN + (size_t)h * NN;
        float acc = 0.f;
#pragma unroll 4
        for (int i = lane; i < NN; i += 32) acc += row[i] * sA[i];
        for (int o = 16; o > 0; o >>= 1) acc += __shfl_xor(acc, o, 32);
        if (lane == 0) sctx[h] = acc;
    }
    __syncthreads();

    // c_ptr[b,t] = sum_k W_ptr[t, 128+k] * context[b,k]
    float cp = 0.f;
    const float* wr = W_ptr + (size_t)t * 256 + 128;
#pragma unroll 8
    for (int k = 0; k < HH; ++k) cp += wr[k] * sctx[k];
    c_ptr[b * HH + t] = cp;
}

// -------------------------------------------------------------------------
// Kernel 5: probs[b,n] = v_ptr . tanh(Wp_s @ s_n + c_ptr[b])  (WMMA f32)
// Same structure as kernel 2, single weight matrix.
// -------------------------------------------------------------------------
__global__ void __launch_bounds__(128)
ptr_scores_kernel(const float* __restrict__ S, const float* __restrict__ W_ptr,
                  const float* __restrict__ v_ptr, const float* __restrict__ c_ptr,
                  float* __restrict__ probs)
{
    __shared__ __align__(16) float sS[32 * 128];
    __shared__ float sC[HH], sV[HH];

    const int b  = blockIdx.y;
    const int n0 = blockIdx.x * 128;
    const int t  = threadIdx.x;
    const int lane = t & 31;
    const int wave = t >> 5;
    const int l15 = lane & 15;
    const int nl0 = (wave << 4) + l15;
    const int nl1 = nl0 + 64;
    const bool hi = lane >= 16;
    const int kofs = hi ? 2 : 0;

    sC[t] = c_ptr[b * HH + t];
    sV[t] = v_ptr[t];
    const size_t sb = (size_t)b * HH * NN;

    v8f acc[2][8];
#pragma unroll
    for (int q = 0; q < 2; ++q)
#pragma unroll
        for (int m = 0; m < 8; ++m)
            acc[q][m] = (v8f){0.f,0.f,0.f,0.f,0.f,0.f,0.f,0.f};

    for (int stage = 0; stage < 4; ++stage) {
        const int kbase = stage * 32;
        __syncthreads();
#pragma unroll
        for (int i = 0; i < 8; ++i) {
            const int e4 = i * 128 + t;
            const int k  = e4 >> 5;
            const int j  = e4 & 31;
            ((float4*)sS)[e4] = *(const float4*)(S + sb + (size_t)(kbase + k) * NN + n0 + j * 4);
        }
        __syncthreads();

        const float* Wk = W_ptr + (size_t)l15 * 256 + kbase + kofs;
        for (int k0 = 0; k0 < 32; k0 += 4) {
            v2f bP0, bP1;
            bP0.x = sS[(k0 + kofs) * 128 + nl0]; bP0.y = sS[(k0 + kofs + 1) * 128 + nl0];
            bP1.x = sS[(k0 + kofs) * 128 + nl1]; bP1.y = sS[(k0 + kofs + 1) * 128 + nl1];
#pragma unroll
            for (int m = 0; m < 8; ++m) {
                const float* Arow = Wk + (size_t)(m * 16) * 256 + k0;
                v2f aP; aP.x = Arow[0]; aP.y = Arow[1];
                acc[0][m] = __builtin_amdgcn_wmma_f32_16x16x4_f32(
                    false, aP, false, bP0, (short)0, acc[0][m], false, false);
                acc[1][m] = __builtin_amdgcn_wmma_f32_16x16x4_f32(
                    false, aP, false, bP1, (short)0, acc[1][m], false, false);
            }
        }
    }

    float p0 = 0.f, p1 = 0.f;
#pragma unroll
    for (int m = 0; m < 8; ++m) {
        const int rbase = m * 16 + (hi ? 8 : 0);
#pragma unroll
        for (int v = 0; v < 8; ++v) {
            const int row = rbase + v;
            const float w = sV[row], cb = sC[row];
            p0 += w * tanhf(acc[0][m][v] + cb);
            p1 += w * tanhf(acc[1][m][v] + cb);
        }
    }
    p0 += __shfl_xor(p0, 16, 32);
    p1 += __shfl_xor(p1, 16, 32);
    if (!hi) {
        probs[(size_t)b * NN + n0 + nl0] = p0;
        probs[(size_t)b * NN + n0 + nl1] = p1;
    }
}

// -------------------------------------------------------------------------
extern "C" void kernel_launch(void* const* d_in, const int* in_sizes, int n_in,
                              void* d_out, int out_size, void* d_ws, size_t ws_size,
                              hipStream_t stream)
{
    (void)in_sizes; (void)n_in; (void)out_size; (void)ws_size;
    const float* static_hidden  = (const float*)d_in[0];
    const float* dynamic_hidden = (const float*)d_in[1];
    const float* decoder_hidden = (const float*)d_in[2];
    const float* last_hh        = (const float*)d_in[3];
    const float* w_ih           = (const float*)d_in[4];
    const float* w_hh           = (const float*)d_in[5];
    const float* b_ih           = (const float*)d_in[6];
    const float* b_hh           = (const float*)d_in[7];
    const float* v_attn         = (const float*)d_in[8];
    const float* W_attn         = (const float*)d_in[9];
    const float* v_ptr          = (const float*)d_in[10];
    const float* W_ptr          = (const float*)d_in[11];

    float* out    = (float*)d_out;
    float* probs  = out;                       // B*N floats
    float* new_hh = out + (size_t)BB * NN;     // B*H floats

    float* ws     = (float*)d_ws;
    float* c_attn = ws;                               // B*H
    float* Ebuf   = ws + (size_t)BB * HH;             // B*N (scores -> attns in place)
    float* c_ptr  = ws + (size_t)BB * HH + (size_t)BB * NN;  // B*H

    gru_kernel<<<BB, 128, 0, stream>>>(decoder_hidden, last_hh, w_ih, w_hh, b_ih, b_hh,
                                       W_attn, new_hh, c_attn);
    attn_scores_kernel<<<dim3(NN / 128, BB), 128, 0, stream>>>(static_hidden, dynamic_hidden,
                                                               W_attn, v_attn, c_attn, Ebuf);
    softmax_kernel<<<BB, 128, 0, stream>>>(Ebuf);
    context_kernel<<<BB, 128, 0, stream>>>(static_hidden, Ebuf, W_ptr, c_ptr);
    ptr_scores_kernel<<<dim3(NN / 128, BB), 128, 0, stream>>>(static_hidden, W_ptr, v_ptr,
                                                              c_ptr, probs);
}